// myDKT_80968723464645
// MI455X (gfx1250) — compile-verified
//
#include <hip/hip_runtime.h>
#include <hip/hip_bf16.h>

#define DIM 128

typedef __attribute__((ext_vector_type(2))) float v2f;
typedef __attribute__((ext_vector_type(8))) float v8f;

// ---------------------------------------------------------------------------
// init: deg[n] = 1.0 (self-loop weight), out = 0 (scatter target)
// ---------------------------------------------------------------------------
__global__ void dkt_init_kernel(float* __restrict__ deg, float* __restrict__ out,
                                int N, int ND) {
  int i = blockIdx.x * blockDim.x + threadIdx.x;
  if (i < ND) out[i] = 0.0f;
  if (i < N)  deg[i] = 1.0f;
}

// ---------------------------------------------------------------------------
// degree: deg[dst[e]] += 1  (non-returning f32 atomic -> global_atomic_add_f32)
// ---------------------------------------------------------------------------
__global__ void dkt_degree_kernel(const int* __restrict__ dst,
                                  float* __restrict__ deg, int E) {
  int e = blockIdx.x * blockDim.x + threadIdx.x;
  if (e < E) unsafeAtomicAdd(&deg[dst[e]], 1.0f);
}

// ---------------------------------------------------------------------------
// h = x @ W_gcn with x = W_embed (row 0 zeroed), via V_WMMA_F32_16X16X4_F32.
// Block = 8 waves = one 16-row strip; wave w owns column tile n0 = 16*w.
// W_gcn (64 KB) is staged once per block into LDS; B-fragments come from
// ds_load. A-fragments are unconditional global_load_b64 of the contiguous
// (k, k+1) pair at a clamped (always in-bounds) row address; row validity is
// resolved at store time (row 0 -> 0.0f, rows >= N -> not stored), so the
// K-loop has no branches and fully unrolls into 32 chained WMMAs.
//
// Fragment layouts per CDNA5 ISA:
//   A 16x4 f32 : lanes 0-15 hold K=k0,k0+1 (M=lane), lanes 16-31 K=k0+2,k0+3
//   B 4x16 f32 : mirrored (N = lane&15)
//   C/D 16x16  : VGPR j holds M = j (lanes 0-15) / j+8 (lanes 16-31)
// ---------------------------------------------------------------------------
__global__ void dkt_gemm_wmma_kernel(const float* __restrict__ W_embed,
                                     const float* __restrict__ W_gcn,
                                     float* __restrict__ h, int N) {
  __shared__ float Bs[DIM * DIM];                      // 64 KB: whole W_gcn

  // cooperative stage: 16384 floats / 256 threads = 16 float4 per thread
  for (int i = threadIdx.x * 4; i < DIM * DIM; i += 256 * 4) {
    *(float4*)(Bs + i) = *(const float4*)(W_gcn + i);
  }
  __syncthreads();

  const int lane = threadIdx.x & 31;
  const int wave = threadIdx.x >> 5;                   // 0..7 -> column tile
  const int m0 = blockIdx.x * 16;
  const int n0 = wave * 16;
  const int mn = lane & 15;
  const int kHalf = (lane >> 4) * 2;                   // 0 or 2

  const int m = m0 + mn;
  const int mSafe = (m < N) ? m : (N - 1);             // always in-bounds
  const float* aPtr = W_embed + (size_t)mSafe * DIM + kHalf;
  const float* bPtr = Bs + kHalf * DIM + n0 + mn;

  v8f acc = {};
  #pragma unroll
  for (int k0 = 0; k0 < DIM; k0 += 4) {
    v2f a = *(const v2f*)(aPtr + k0);                  // global_load_b64
    v2f b;
    b.x = bPtr[(size_t)k0 * DIM];                      // ds_load (2addr pair)
    b.y = bPtr[(size_t)k0 * DIM + DIM];
    // (neg_a, A, neg_b, B, c_mod, C, reuse_a, reuse_b)
    acc = __builtin_amdgcn_wmma_f32_16x16x4_f32(false, a, false, b,
                                                (short)0, acc, false, false);
  }

  const int rowBase = m0 + ((lane >> 4) << 3);         // +8 for upper half-wave
  #pragma unroll
  for (int j = 0; j < 8; ++j) {
    const int row = rowBase + j;
    if (row < N) {
      // embedding row 0 is zeroed in the reference -> h[0,:] = 0 exactly
      h[(size_t)row * DIM + n0 + mn] = (row == 0) ? 0.0f : acc[j];
    }
  }
}

// ---------------------------------------------------------------------------
// scatter: out[dst] += h[src] * rsqrt(deg[src]*deg[dst]).
// One edge per wave32; each lane owns a float4 (32*4 = 128 dims).
// h and out are L2-resident (25.6 MB each vs 192 MB L2), so this is
// L2-atomic-throughput bound; non-returning global_atomic_add_f32.
// ---------------------------------------------------------------------------
__global__ void dkt_scatter_kernel(const int* __restrict__ src,
                                   const int* __restrict__ dst,
                                   const float* __restrict__ deg,
                                   const float* __restrict__ h,
                                   float* __restrict__ out, int E) {
  const int lane = threadIdx.x & 31;
  const int e = blockIdx.x * (blockDim.x >> 5) + (threadIdx.x >> 5);
  if (e >= E) return;
  const int s = src[e];
  const int t = dst[e];
  const float norm = rsqrtf(deg[s] * deg[t]);
  const float4 v = *(const float4*)(h + (size_t)s * DIM + lane * 4);
  float* o = out + (size_t)t * DIM + lane * 4;
  unsafeAtomicAdd(o + 0, v.x * norm);
  unsafeAtomicAdd(o + 1, v.y * norm);
  unsafeAtomicAdd(o + 2, v.z * norm);
  unsafeAtomicAdd(o + 3, v.w * norm);
}

// ---------------------------------------------------------------------------
// finalize: out += h * (1/deg)  (self-loop term) + bias
// ---------------------------------------------------------------------------
__global__ void dkt_finalize_kernel(const float* __restrict__ h,
                                    const float* __restrict__ deg,
                                    const float* __restrict__ bias,
                                    float* __restrict__ out, int ND) {
  int i = blockIdx.x * blockDim.x + threadIdx.x;
  if (i >= ND) return;
  const int n = i >> 7;      // / 128
  const int d = i & 127;     // % 128
  out[i] += h[i] * (1.0f / deg[n]) + bias[d];
}

// ---------------------------------------------------------------------------
extern "C" void kernel_launch(void* const* d_in, const int* in_sizes, int n_in,
                              void* d_out, int out_size, void* d_ws, size_t ws_size,
                              hipStream_t stream) {
  const float* W_embed = (const float*)d_in[0];   // [N, 128]
  const float* W_gcn   = (const float*)d_in[1];   // [128, 128]
  const float* b_gcn   = (const float*)d_in[2];   // [128]
  const int*   edge    = (const int*)d_in[3];     // [2, E] flat (src row, dst row)
  // d_in[4], d_in[5] (e_inputs, a_inputs) are unused by the reference output.

  const int N  = in_sizes[0] / DIM;
  const int E  = in_sizes[3] / 2;
  const int ND = N * DIM;

  // workspace layout: [deg: N floats][pad to 64][h: N*DIM floats]
  float* deg = (float*)d_ws;
  const size_t hOff = ((size_t)N + 63) & ~(size_t)63;
  float* h = (float*)d_ws + hOff;
  float* out = (float*)d_out;

  const int* srcIdx = edge;
  const int* dstIdx = edge + E;

  dkt_init_kernel<<<(ND + 255) / 256, 256, 0, stream>>>(deg, out, N, ND);
  dkt_degree_kernel<<<(E + 255) / 256, 256, 0, stream>>>(dstIdx, deg, E);

  const int tilesM = (N + 15) / 16;
  dkt_gemm_wmma_kernel<<<tilesM, 256, 0, stream>>>(W_embed, W_gcn, h, N);

  dkt_scatter_kernel<<<(E + 7) / 8, 256, 0, stream>>>(srcIdx, dstIdx, deg, h, out, E);
  dkt_finalize_kernel<<<(ND + 255) / 256, 256, 0, stream>>>(h, deg, b_gcn, out, ND);
}